// TemporalConsistencyLoss_55825984913846
// MI455X (gfx1250) — compile-verified
//
#include <hip/hip_runtime.h>

typedef __attribute__((ext_vector_type(2))) float v2f;
typedef __attribute__((ext_vector_type(8))) float v8f;

#define LOSS_WEIGHT 0.1f
#define MAX_FRAMES_GAP 5

// ---------------------------------------------------------------------------
// Kernel 0: zero the two global accumulators (loss, count) in workspace.
// ---------------------------------------------------------------------------
__global__ void tcl_init_kernel(float* acc) {
    acc[0] = 0.0f;
    acc[1] = 0.0f;
}

// ---------------------------------------------------------------------------
// Kernel 1: row norms. One wave (32 lanes) per row; shuffle reduction.
// ---------------------------------------------------------------------------
__global__ void tcl_norms_kernel(const float* __restrict__ E,
                                 float* __restrict__ norms,
                                 int B, int D) {
    const int wave = threadIdx.x >> 5;
    const int lane = threadIdx.x & 31;
    const int row  = blockIdx.x * (blockDim.x >> 5) + wave;
    if (row >= B) return;
    const float* p = E + (size_t)row * (size_t)D;
    float s = 0.0f;
    for (int t = lane; t < D; t += 32) {
        float v = p[t];
        s += v * v;
    }
    #pragma unroll
    for (int off = 16; off > 0; off >>= 1) s += __shfl_xor(s, off, 32);
    if (lane == 0) norms[row] = s;
}

// ---------------------------------------------------------------------------
// Kernel 2: one wave per 16x16 (i,j) tile of the pair matrix.
//   - integer mask precheck (wave-uniform early exit keeps EXEC==all-1s)
//   - active tiles compute the Gram block with V_WMMA_F32_16X16X4_F32
//   - masked weighted accumulation, shuffle reduce, 2 global float atomics
// ---------------------------------------------------------------------------
__global__ __launch_bounds__(32)
void tcl_tile_kernel(const float* __restrict__ E,
                     const int*   __restrict__ frames,
                     const int*   __restrict__ vids,
                     const float* __restrict__ norms,
                     float*       __restrict__ acc,
                     int B, int D) {
    const int tj = blockIdx.x;   // tile column (j)
    const int ti = blockIdx.y;   // tile row    (i)
    if (tj < ti) return;         // strictly-lower tiles have no j>i pairs

    const int i0   = ti * 16;
    const int j0   = tj * 16;
    const int lane = threadIdx.x;        // 0..31 (wave32)
    const int l16  = lane & 15;
    const int half = lane >> 4;          // 0 or 1

    // Per-lane metadata: lane l16 caches row (i0+l16) / (j0+l16) info.
    const int fi_self = frames[i0 + l16];
    const int vi_self = vids[i0 + l16];
    const int fj      = frames[j0 + l16];
    const int vj      = vids[j0 + l16];

    // ---- mask precheck: does ANY pair in this 16x16 tile survive? ----
    bool act = false;
    #pragma unroll
    for (int r = 0; r < 8; ++r) {
        const int m  = r + 8 * half;                // C-layout row for this lane
        const int fi = __shfl(fi_self, m, 32);
        const int vi = __shfl(vi_self, m, 32);
        const int df = abs(fi - fj);
        act |= (vi == vj) && (df <= MAX_FRAMES_GAP) && ((j0 + l16) > (i0 + m));
    }
    if (!__any(act)) return;     // wave-uniform: EXEC stays all-1s for WMMA

    // ---- Gram block via V_WMMA_F32_16X16X4_F32 ----
    // A (16x4, MxK): lane holds M=l16, VGPR0/1 = K = 2*half, 2*half+1
    // B (4x16, KxN): lane holds N=l16, VGPR0/1 = K = 2*half, 2*half+1
    // Both are contiguous 8-byte (float2) loads from row-major E.
    const float* arow = E + (size_t)(i0 + l16) * (size_t)D + 2 * half;
    const float* brow = E + (size_t)(j0 + l16) * (size_t)D + 2 * half;

    v8f c = {};
    #pragma unroll 4
    for (int k = 0; k < D; k += 4) {
        v2f a = *(const v2f*)(arow + k);
        v2f b = *(const v2f*)(brow + k);
        c = __builtin_amdgcn_wmma_f32_16x16x4_f32(
                /*neg_a=*/false, a, /*neg_b=*/false, b,
                /*c_mod=*/(short)0, c, /*reuse_a=*/false, /*reuse_b=*/false);
    }

    // ---- masked, weighted accumulation over the 8 C rows this lane owns ----
    const float ni_self = norms[i0 + l16];
    const float nj      = norms[j0 + l16];
    const float invD    = 1.0f / (float)D;

    float loss = 0.0f;
    float cnt  = 0.0f;
    #pragma unroll
    for (int r = 0; r < 8; ++r) {
        const int m  = r + 8 * half;
        const int fi = __shfl(fi_self, m, 32);
        const int vi = __shfl(vi_self, m, 32);
        const float ni = __shfl(ni_self, m, 32);
        const int df = abs(fi - fj);
        const bool msk = (vi == vj) && (df <= MAX_FRAMES_GAP) &&
                         ((j0 + l16) > (i0 + m));
        if (msk) {
            const float sq = fmaxf(ni + nj - 2.0f * c[r], 0.0f);
            const float w  = 1.0f / (1.0f + (float)df);
            loss += w * sq * invD;
            cnt  += 1.0f;
        }
    }

    // ---- wave reduction + global float atomics ----
    #pragma unroll
    for (int off = 16; off > 0; off >>= 1) {
        loss += __shfl_xor(loss, off, 32);
        cnt  += __shfl_xor(cnt,  off, 32);
    }
    if (lane == 0) {
        atomicAdd(&acc[0], loss);
        atomicAdd(&acc[1], cnt);
    }
}

// ---------------------------------------------------------------------------
// Kernel 3: finalize scalar output.
// ---------------------------------------------------------------------------
__global__ void tcl_finalize_kernel(const float* acc, float* out) {
    out[0] = LOSS_WEIGHT * acc[0] / fmaxf(acc[1], 1.0f);
}

// ---------------------------------------------------------------------------
extern "C" void kernel_launch(void* const* d_in, const int* in_sizes, int n_in,
                              void* d_out, int out_size, void* d_ws, size_t ws_size,
                              hipStream_t stream) {
    const float* E      = (const float*)d_in[0];   // embeddings (B, D) fp32
    const int*   frames = (const int*)d_in[1];     // frame_numbers (B,)
    const int*   vids   = (const int*)d_in[2];     // video_ids (B,)
    float*       out    = (float*)d_out;           // scalar fp32

    const int BD = in_sizes[0];
    const int B  = in_sizes[1];
    const int D  = BD / B;                         // 256

    // Workspace layout: [0..B) norms, [B] loss accum, [B+1] count accum.
    float* norms = (float*)d_ws;
    float* acc   = norms + B;

    tcl_init_kernel<<<1, 1, 0, stream>>>(acc);

    const int wavesPerBlock = 8;                   // 256 threads = 8 waves
    const int normBlocks = (B + wavesPerBlock - 1) / wavesPerBlock;
    tcl_norms_kernel<<<normBlocks, 32 * wavesPerBlock, 0, stream>>>(E, norms, B, D);

    dim3 grid(B / 16, B / 16);                     // (tile_j, tile_i)
    tcl_tile_kernel<<<grid, 32, 0, stream>>>(E, frames, vids, norms, acc, B, D);

    tcl_finalize_kernel<<<1, 1, 0, stream>>>(acc, out);
}